// GCN_20641612824768
// MI455X (gfx1250) — compile-verified
//
#include <hip/hip_runtime.h>
#include <hip/hip_bf16.h>

// ---------------------------------------------------------------------------
// GCN forward on MI455X (gfx1250): bandwidth-bound edge scatter (prefetched,
// grid-stride) + fp32 WMMA GEMMs (V_WMMA_F32_16X16X4_F32, 16x32 tile/wave).
// ---------------------------------------------------------------------------

typedef __attribute__((ext_vector_type(2))) float v2f;
typedef __attribute__((ext_vector_type(8))) float v8f;

#define LN_EPS 1e-5f

// ---------------- utility ----------------
__global__ void zero_f32_k(float* __restrict__ p, long long n) {
    long long i = (long long)blockIdx.x * blockDim.x + threadIdx.x;
    long long stride = (long long)gridDim.x * blockDim.x;
    for (; i < n; i += stride) p[i] = 0.0f;
}

// ---------------- degrees ----------------
__global__ void degree_k(const int* __restrict__ src, const int* __restrict__ dst,
                         float* __restrict__ outdeg, float* __restrict__ indeg, int E) {
    int e = blockIdx.x * blockDim.x + threadIdx.x;
    if (e < E) {
        atomicAdd(&outdeg[src[e]], 1.0f);
        atomicAdd(&indeg[dst[e]], 1.0f);
    }
}

__global__ void invsqrt_k(float* __restrict__ d, int n) {
    int i = blockIdx.x * blockDim.x + threadIdx.x;
    if (i < n) d[i] = rsqrtf(fmaxf(d[i], 1.0f));
}

// coef[e] = max(w[e,0..3]) * out_inv[src[e]]  (folds source-side norm into edge)
__global__ void coef_k(const float* __restrict__ w4, const int* __restrict__ src,
                       const float* __restrict__ out_inv, float* __restrict__ coef, int E) {
    int e = blockIdx.x * blockDim.x + threadIdx.x;
    if (e < E) {
        float4 q = ((const float4*)w4)[e];
        float m = fmaxf(fmaxf(q.x, q.y), fmaxf(q.z, q.w));
        coef[e] = m * out_inv[src[e]];
    }
}

// ---------------- edge gather + atomic scatter-add ----------------
// Grid-stride, one wave per edge per step; lane handles dims lane, lane+32,...
// -> coalesced 128B loads and f32 atomic adds (agg buffer is L2-resident).
// Prefetch the NEXT edge's source row (global_prefetch_b8) to hide the random
// row-gather latency behind the current edge's atomics.
__global__ void scatter_k(const float* __restrict__ X, const int* __restrict__ src,
                          const int* __restrict__ dst, const float* __restrict__ coef,
                          float* __restrict__ agg, int D, int E) {
    int waves_per_blk = blockDim.x >> 5;
    int wv = blockIdx.x * waves_per_blk + (threadIdx.x >> 5);
    int total_waves = gridDim.x * waves_per_blk;
    int lane = threadIdx.x & 31;

    for (int e = wv; e < E; e += total_waves) {
        int s = src[e], d = dst[e];
        float cf = coef[e];
        const float* xs = X + (size_t)s * D;
        float* ad = agg + (size_t)d * D;

        int en = e + total_waves;                 // prefetch next edge's source row
        if (en < E) {
            const float* xn = X + (size_t)src[en] * D;
            if (lane * 32 < D)                    // one lane per 128B cacheline
                __builtin_prefetch(xn + lane * 32, 0, 0);
        }

        for (int j = lane; j < D; j += 32)
            atomicAdd(&ad[j], xs[j] * cf);
    }
}

// ---------------- fp32 WMMA GEMM:  Y = act((A .* rowscale) @ W) ----------------
// One wave computes a 16x32 output tile (two N-tiles sharing one A fragment)
// via K/4 pairs of V_WMMA_F32_16X16X4_F32. A: [M,K] row-major, W: [K,Nout]
// row-major, Y row stride ldY (column-offset via pre-offset pointer).
// Wave-uniform guard keeps EXEC all-ones at every WMMA (ISA §7.12 requirement).
__global__ void wmma_gemm_k(const float* __restrict__ A, const float* __restrict__ rowscale,
                            const float* __restrict__ W, float* __restrict__ Y,
                            int ldY, int M, int K, int Nout, int do_relu) {
    int waves_per_blk = blockDim.x >> 5;
    int gw = blockIdx.x * waves_per_blk + (threadIdx.x >> 5);
    int ntiles2 = Nout >> 5;                     // 32-wide N tiles
    int mtiles = M >> 4;
    if (gw >= mtiles * ntiles2) return;          // wave-uniform

    int tm = gw / ntiles2;
    int tn = gw - tm * ntiles2;
    int lane = threadIdx.x & 31;
    int lo = lane & 15;                          // M (A) / N (B,C) index within tile
    int hi = lane >> 4;                          // K-half selector for A/B fragments

    int row = (tm << 4) + lo;
    int col0 = (tn << 5) + lo;                   // first 16-col tile
    int col1 = col0 + 16;                        // second 16-col tile
    float s = rowscale ? rowscale[row] : 1.0f;   // fused D_in^{-1/2} row scaling
    const float* Arow = A + (size_t)row * K;

    v8f c0 = {};
    v8f c1 = {};
    for (int kk = 0; kk < K; kk += 4) {
        int ka = kk + (hi << 1);                 // lanes 0-15: K=kk,kk+1 ; 16-31: kk+2,kk+3
        v2f a = *(const v2f*)(Arow + ka);        // 8B-aligned (ka even, K mult of 4)
        a.x *= s;
        a.y *= s;
        const float* Wk0 = W + (size_t)ka * Nout;
        const float* Wk1 = W + (size_t)(ka + 1) * Nout;
        v2f b0, b1;
        b0.x = Wk0[col0]; b0.y = Wk1[col0];
        b1.x = Wk0[col1]; b1.y = Wk1[col1];
        c0 = __builtin_amdgcn_wmma_f32_16x16x4_f32(false, a, false, b0,
                                                   (short)0, c0, false, false);
        c1 = __builtin_amdgcn_wmma_f32_16x16x4_f32(false, a, false, b1,
                                                   (short)0, c1, false, false);
    }

    int rbase = (tm << 4) + (hi << 3);           // C layout: VGPR i -> M = i + 8*hi
#pragma unroll
    for (int i = 0; i < 8; i++) {
        float v0 = c0[i], v1 = c1[i];
        if (do_relu) { v0 = fmaxf(v0, 0.0f); v1 = fmaxf(v1, 0.0f); }
        size_t rb = (size_t)(rbase + i) * ldY;
        Y[rb + col0] = v0;
        Y[rb + col1] = v1;
    }
}

// ---------------- LayerNorm(256) + ReLU, one wave per row ----------------
__global__ void ln_relu_k(float* __restrict__ f1, int ld,
                          const float* __restrict__ gamma, const float* __restrict__ beta,
                          int M) {
    int row = blockIdx.x * (blockDim.x >> 5) + (threadIdx.x >> 5);
    if (row >= M) return;
    int lane = threadIdx.x & 31;
    float* p = f1 + (size_t)row * ld;
    float v[8];
    float sum = 0.0f;
#pragma unroll
    for (int j = 0; j < 8; j++) { v[j] = p[lane + 32 * j]; sum += v[j]; }
#pragma unroll
    for (int m = 16; m; m >>= 1) sum += __shfl_xor(sum, m, 32);
    float mu = sum * (1.0f / 256.0f);
    float var = 0.0f;
#pragma unroll
    for (int j = 0; j < 8; j++) { float d = v[j] - mu; var += d * d; }
#pragma unroll
    for (int m = 16; m; m >>= 1) var += __shfl_xor(var, m, 32);
    float rs = rsqrtf(var * (1.0f / 256.0f) + LN_EPS);
#pragma unroll
    for (int j = 0; j < 8; j++) {
        int cix = lane + 32 * j;
        float y = (v[j] - mu) * rs * gamma[cix] + beta[cix];
        p[cix] = fmaxf(y, 0.0f);
    }
}

// ---------------- readout: segment_sum over graph ids ----------------
__global__ void readout_k(const float* __restrict__ x3, const int* __restrict__ gid,
                          float* __restrict__ out) {
    int n = blockIdx.x;
    int d = threadIdx.x;                         // 128 threads = OUT_DIM
    atomicAdd(&out[(size_t)gid[n] * 128 + d], x3[(size_t)n * 128 + d]);
}

__global__ void l2norm_k(float* __restrict__ out) {
    __shared__ float sh[128];
    int g = blockIdx.x;
    float v = out[(size_t)g * 128 + threadIdx.x];
    sh[threadIdx.x] = v * v;
    __syncthreads();
    for (int s = 64; s > 0; s >>= 1) {
        if (threadIdx.x < s) sh[threadIdx.x] += sh[threadIdx.x + s];
        __syncthreads();
    }
    float norm = fmaxf(sqrtf(sh[0]), 1e-12f);
    out[(size_t)g * 128 + threadIdx.x] = v / norm;
}

// ---------------------------------------------------------------------------
static inline int cdiv(int a, int b) { return (a + b - 1) / b; }

extern "C" void kernel_launch(void* const* d_in, const int* in_sizes, int n_in,
                              void* d_out, int out_size, void* d_ws, size_t ws_size,
                              hipStream_t stream) {
    const float* x     = (const float*)d_in[0];
    const float* w     = (const float*)d_in[1];
    const float* W1    = (const float*)d_in[2];
    const float* fc1W  = (const float*)d_in[3];
    const float* gamma = (const float*)d_in[4];
    const float* beta  = (const float*)d_in[5];
    const float* W2    = (const float*)d_in[6];
    const float* W3    = (const float*)d_in[7];
    const int*   src   = (const int*)d_in[8];
    const int*   dst   = (const int*)d_in[9];
    const int*   gid   = (const int*)d_in[10];
    float* out = (float*)d_out;

    const int E = in_sizes[8];       // 320000
    const int N = in_sizes[10];      // 40000
    const int IN = 128, HID = 256, CAT = 512, OUT = 128;

    // workspace layout (floats)
    float* wf      = (float*)d_ws;
    float* coef    = wf;                          // E
    float* out_inv = coef + E;                    // N
    float* in_inv  = out_inv + N;                 // N
    float* agg     = in_inv + N;                  // N*512 (reused per conv)
    float* cat     = agg + (size_t)N * CAT;       // N*512  = [x1 | f1]
    float* x2      = cat + (size_t)N * CAT;       // N*256
    float* x3      = x2 + (size_t)N * HID;        // N*128

    const int ZB = 2048, ZT = 256;
    const int SCAT_BLOCKS = 4096;                 // grid-stride edge waves

    // 1) degrees -> D^{-1/2}
    zero_f32_k<<<ZB, ZT, 0, stream>>>(out_inv, (long long)2 * N);
    degree_k<<<cdiv(E, 256), 256, 0, stream>>>(src, dst, out_inv, in_inv, E);
    invsqrt_k<<<cdiv(2 * N, 256), 256, 0, stream>>>(out_inv, 2 * N);

    // 2) per-edge coefficient
    coef_k<<<cdiv(E, 256), 256, 0, stream>>>(w, src, out_inv, coef, E);

    // 3) conv1: agg = scatter(x * coef) ; x1 = relu((agg .* in_inv) @ W1) -> cat[:, :256]
    zero_f32_k<<<ZB, ZT, 0, stream>>>(agg, (long long)N * IN);
    scatter_k<<<SCAT_BLOCKS, 256, 0, stream>>>(x, src, dst, coef, agg, IN, E);
    {
        int waves = (N >> 4) * (HID >> 5);
        wmma_gemm_k<<<cdiv(waves, 8), 256, 0, stream>>>(agg, in_inv, W1, cat, CAT,
                                                        N, IN, HID, /*relu=*/1);
    }

    // 4) fc1: f1 = relu(LN(x @ fc1_W)) -> cat[:, 256:]
    {
        int waves = (N >> 4) * (HID >> 5);
        wmma_gemm_k<<<cdiv(waves, 8), 256, 0, stream>>>(x, nullptr, fc1W, cat + HID, CAT,
                                                        N, IN, HID, /*relu=*/0);
    }
    ln_relu_k<<<cdiv(N, 8), 256, 0, stream>>>(cat + HID, CAT, gamma, beta, N);

    // 5) conv2: K=512 -> x2 [N,256]
    zero_f32_k<<<ZB, ZT, 0, stream>>>(agg, (long long)N * CAT);
    scatter_k<<<SCAT_BLOCKS, 256, 0, stream>>>(cat, src, dst, coef, agg, CAT, E);
    {
        int waves = (N >> 4) * (HID >> 5);
        wmma_gemm_k<<<cdiv(waves, 8), 256, 0, stream>>>(agg, in_inv, W2, x2, HID,
                                                        N, CAT, HID, /*relu=*/1);
    }

    // 6) conv3: K=256 -> x3 [N,128]
    zero_f32_k<<<ZB, ZT, 0, stream>>>(agg, (long long)N * HID);
    scatter_k<<<SCAT_BLOCKS, 256, 0, stream>>>(x2, src, dst, coef, agg, HID, E);
    {
        int waves = (N >> 4) * (OUT >> 5);
        wmma_gemm_k<<<cdiv(waves, 8), 256, 0, stream>>>(agg, in_inv, W3, x3, OUT,
                                                        N, HID, OUT, /*relu=*/1);
    }

    // 7) readout + row-wise L2 normalize
    zero_f32_k<<<cdiv(64 * OUT, 256), 256, 0, stream>>>(out, 64 * OUT);
    readout_k<<<N, OUT, 0, stream>>>(x3, gid, out);
    l2norm_k<<<64, OUT, 0, stream>>>(out);
}